// GCN_76682346102827
// MI455X (gfx1250) — compile-verified
//
#include <hip/hip_runtime.h>

typedef float v2f __attribute__((ext_vector_type(2)));
typedef float v8f __attribute__((ext_vector_type(8)));

// ---------------------------------------------------------------------------
// Dense GEMM via V_WMMA_F32_16X16X4_F32, LDS-staged.
//   C[M,N] = act( A[M,K] @ B[K,N] + bias[N] )
// Compile-time K, N => all strides are immediates.
// Block = 256 threads = 8 waves. Block tile: 128(M) x 64(N); each wave owns a
// 16x64 strip = four 16x16 WMMA tiles. K is processed in 64-wide chunks
// staged through LDS:
//   ldsA: 128 rows x 64 (stride 68: lane fragment b64 reads are conflict-free)
//   ldsB: paired-K layout, pair p=(2p,2p+1) at stride 160 so the b64 fragment
//         read puts lanes 0-15 and 16-31 in disjoint 32-bank halves.
// WMMA fragment mapping (ISA 7.12.2):
//   A: lane = m + 16*(k/2 half), holds A[m][2h..2h+1]  -> one b64 read
//   B: lane = n + 16*half,       holds B[2h..2h+1][n]  -> one b64 read
//   D: vgpr i, lane l -> row i + 8*(l>>4), col (l&15)
// ---------------------------------------------------------------------------
template<int K, int N, bool RELU_IN, bool RELU_OUT>
__global__ __launch_bounds__(256)
void gcn_gemm_wmma(const float* __restrict__ A,
                   const float* __restrict__ Bg,
                   const float* __restrict__ bias,
                   float* __restrict__ C,
                   int M)
{
    constexpr int KC   = 64;    // K chunk
    constexpr int APAD = 68;    // ldsA row stride (floats)
    constexpr int PS   = 160;   // ldsB pair-row stride (floats), 160 % 64 == 32

    __shared__ float ldsA[128 * APAD];        // 34.0 KB
    __shared__ float ldsB[(KC / 2) * PS];     // 20.0 KB

    const int tid  = threadIdx.x;
    const int lane = tid & 31;
    const int wave = tid >> 5;
    const int l15  = lane & 15;
    const int kh   = lane >> 4;
    const int m0   = blockIdx.x * 128;
    const int n0   = blockIdx.y * 64;

    v8f acc[4];
    const v8f vzero = {0.f, 0.f, 0.f, 0.f, 0.f, 0.f, 0.f, 0.f};
#pragma unroll
    for (int t = 0; t < 4; ++t) acc[t] = vzero;

    // Cooperative A-load mapping: thread = (row ar, half ah); 32 contiguous
    // floats per thread per chunk (8 x float4).
    const int ar  = tid >> 1;
    const int ah  = tid & 1;
    int arow = m0 + ar;
    if (arow >= M) arow = M - 1;              // clamp for edge block
    const float* Agp    = A + (size_t)arow * K + ah * 32;
    float*       ldsArp = &ldsA[ar * APAD + ah * 32];

    // Cooperative B-load mapping: thread = (col bn, pair-group bp).
    const int bn = tid & 63;
    const int bp = tid >> 6;

    const float* aFrag = &ldsA[(wave * 16 + l15) * APAD + 2 * kh];

#pragma unroll 1
    for (int k0 = 0; k0 < K; k0 += KC) {
        // ---- stage A chunk (with fused input-ReLU) ----
#pragma unroll
        for (int j = 0; j < 8; ++j) {
            float4 v = *(const float4*)(Agp + k0 + j * 4);
            if (RELU_IN) {
                v.x = fmaxf(v.x, 0.f); v.y = fmaxf(v.y, 0.f);
                v.z = fmaxf(v.z, 0.f); v.w = fmaxf(v.w, 0.f);
            }
            *(float4*)(ldsArp + j * 4) = v;
        }
        // ---- stage B chunk in paired-K layout ----
#pragma unroll
        for (int it = 0; it < 8; ++it) {
            const int p  = bp + it * 4;           // pair index 0..31
            const int kk = k0 + 2 * p;
            v2f b;
            b.x = Bg[(size_t)kk * N + n0 + bn];
            b.y = Bg[(size_t)(kk + 1) * N + n0 + bn];
            *(v2f*)&ldsB[p * PS + bn * 2] = b;
        }
        if (k0 + KC < K) {                        // prefetch next chunk
            __builtin_prefetch(Agp + k0 + KC, 0, 3);
            __builtin_prefetch(Bg + (size_t)(k0 + KC) * N + n0 + bn, 0, 3);
        }
        __syncthreads();

        // ---- compute: 16 k-steps x 4 WMMA from LDS ----
#pragma unroll
        for (int k = 0; k < KC; k += 4) {
            const v2f a  = *(const v2f*)(aFrag + k);
            const int pr = (k >> 1) + kh;
#pragma unroll
            for (int t = 0; t < 4; ++t) {
                const v2f b = *(const v2f*)&ldsB[pr * PS + (t * 16 + l15) * 2];
                acc[t] = __builtin_amdgcn_wmma_f32_16x16x4_f32(
                    false, a, false, b, (short)0, acc[t], false, false);
            }
        }
        __syncthreads();
    }

    // ---- epilogue: bias (+ReLU) and guarded stores ----
#pragma unroll
    for (int t = 0; t < 4; ++t) {
        const int   n  = n0 + t * 16 + l15;
        const float bv = bias[n];
#pragma unroll
        for (int i = 0; i < 8; ++i) {
            const int r = m0 + wave * 16 + i + 8 * kh;
            if (r < M) {
                float v = acc[t][i] + bv;
                if (RELU_OUT) v = fmaxf(v, 0.f);
                C[(size_t)r * N + n] = v;
            }
        }
    }
}

// ---------------------------------------------------------------------------
// Edge-parallel COO SpMM scatter: y[rows[e], :] += vals[e] * x[cols[e], :]
// Each lane loads one edge (coalesced), broadcasts r/c/v via __shfl (uniform
// source lane -> readlane), then the wave processes the edge with per-lane
// contiguous F/32-float segments: b64/b128 row reads from L2 and per-lane
// global_atomic_add_f32 for the segment-sum.
// ---------------------------------------------------------------------------
template<int F>
__global__ __launch_bounds__(256)
void gcn_spmm_scatter(const int* __restrict__ rows,
                      const int* __restrict__ cols,
                      const float* __restrict__ vals,
                      const float* __restrict__ x,
                      float* __restrict__ y,
                      int n_edges)
{
    constexpr int VEC = F / 32;                  // 8 (F=256) or 2 (F=64)
    const int lane   = threadIdx.x & 31;
    const int wid    = (blockIdx.x * blockDim.x + threadIdx.x) >> 5;
    const int nwaves = (gridDim.x * blockDim.x) >> 5;

    for (int base = wid * 32; base < n_edges; base += nwaves * 32) {
        const int  e  = base + lane;
        const bool ok = e < n_edges;
        const int   r = ok ? rows[e] : 0;
        const int   c = ok ? cols[e] : 0;
        const float v = ok ? vals[e] : 0.0f;

        int cnt = n_edges - base;
        if (cnt > 32) cnt = 32;
        for (int i = 0; i < cnt; ++i) {
            const int   ri = __shfl(r, i, 32);
            const int   ci = __shfl(c, i, 32);
            const float vi = __shfl(v, i, 32);
            const float* xs = x + (size_t)ci * F + lane * VEC;
            float*       yd = y + (size_t)ri * F + lane * VEC;
            float b[VEC];
#pragma unroll
            for (int j = 0; j < VEC; ++j) b[j] = xs[j];
#pragma unroll
            for (int j = 0; j < VEC; ++j) atomicAdd(&yd[j], vi * b[j]);
        }
    }
}

__global__ __launch_bounds__(256)
void gcn_zero_f32(float4* __restrict__ p, size_t n4)
{
    size_t i      = (size_t)blockIdx.x * blockDim.x + threadIdx.x;
    size_t stride = (size_t)gridDim.x * blockDim.x;
    const float4 z = make_float4(0.f, 0.f, 0.f, 0.f);
    for (; i < n4; i += stride) p[i] = z;
}

// ---------------------------------------------------------------------------
// Orchestration.
// Inputs (setup_inputs order):
//  0:x_a[60000,256] 1:x_b[40000,128] 2:rows[3.2M] 3:cols[3.2M] 4:vals[3.2M]
//  5:W1a[256,256] 6:b1a[256] 7:W1b[128,256] 8:b1b[256]
//  9:Wf[256,256] 10:bf[256] 11:W2[256,64] 12:b2[64]
// Output: [100000, 64] f32.
// Workspace: B0, B1 of 100000*256 f32 each (ping-pong: h->t in B0, z->u in B1).
// ---------------------------------------------------------------------------
extern "C" void kernel_launch(void* const* d_in, const int* in_sizes, int n_in,
                              void* d_out, int out_size, void* d_ws, size_t ws_size,
                              hipStream_t stream)
{
    const float* x_a  = (const float*)d_in[0];
    const float* x_b  = (const float*)d_in[1];
    const int*   rows = (const int*)  d_in[2];
    const int*   cols = (const int*)  d_in[3];
    const float* vals = (const float*)d_in[4];
    const float* W1a  = (const float*)d_in[5];
    const float* b1a  = (const float*)d_in[6];
    const float* W1b  = (const float*)d_in[7];
    const float* b1b  = (const float*)d_in[8];
    const float* Wf   = (const float*)d_in[9];
    const float* bf   = (const float*)d_in[10];
    const float* W2   = (const float*)d_in[11];
    const float* b2   = (const float*)d_in[12];
    float*       out  = (float*)d_out;

    const int NN = 100000, MA = 60000, MB = 40000;
    const int NE = 3200000;
    const int H  = 256, NO = 64;

    float* B0 = (float*)d_ws;                // h, then t (SpMM1 output)
    float* B1 = B0 + (size_t)NN * H;         // z, then u (proj output)

    const dim3 blk(256);

    // 1) h = relu(x @ W1 + b1), per node type, concatenated in B0.
    gcn_gemm_wmma<256, 256, false, true>
        <<<dim3((MA + 127) / 128, 4), blk, 0, stream>>>(x_a, W1a, b1a, B0, MA);
    gcn_gemm_wmma<128, 256, false, true>
        <<<dim3((MB + 127) / 128, 4), blk, 0, stream>>>(x_b, W1b, b1b,
                                                        B0 + (size_t)MA * H, MB);

    // 2) z = h @ Wf + bf
    gcn_gemm_wmma<256, 256, false, false>
        <<<dim3((NN + 127) / 128, 4), blk, 0, stream>>>(B0, Wf, bf, B1, NN);

    // 3) t = segment_sum(vals * z[cols])  (B0 reused; h is dead)
    gcn_zero_f32<<<2048, blk, 0, stream>>>((float4*)B0, (size_t)NN * H / 4);
    gcn_spmm_scatter<256><<<2048, blk, 0, stream>>>(rows, cols, vals, B1, B0, NE);

    // 4) u = relu(t) @ W2 + b2  (ReLU fused into A staging; B1 reused, z dead)
    gcn_gemm_wmma<256, 64, true, false>
        <<<dim3((NN + 127) / 128, 1), blk, 0, stream>>>(B0, W2, b2, B1, NN);

    // 5) out = segment_sum(vals * u[cols])
    gcn_zero_f32<<<1024, blk, 0, stream>>>((float4*)out, (size_t)NN * NO / 4);
    gcn_spmm_scatter<64><<<2048, blk, 0, stream>>>(rows, cols, vals, B1, out, NE);
}